// Memory_47734266528128
// MI455X (gfx1250) — compile-verified
//
#include <hip/hip_runtime.h>
#include <cstdint>

typedef __attribute__((ext_vector_type(2))) float v2f;
typedef __attribute__((ext_vector_type(8))) float v8f;
typedef __attribute__((ext_vector_type(4))) unsigned int u32x4;
typedef __attribute__((ext_vector_type(8))) int i32x8;
typedef __attribute__((ext_vector_type(4))) int i32x4;

#define FEAT   256          // K
#define BQ     128          // batch (queries)
#define KC     64           // K-chunk staged per TDM transfer
#define NCHUNK (FEAT / KC)  // 4
#define TN     128          // Mem rows per workgroup tile
#define RS     66           // padded LDS row stride (dwords): 64 + 2-dword TDM pad
#define NBINS  4096         // width-1.0 linear bins over distance range
#define CAP    2048         // boundary-bin capture buffer per row
#define HCHUNK 65536        // elements per histogram/collect workgroup

// ---------------------------------------------------------------- TDM issue
// 2-D tensor descriptor: dim0 = K (stride 256 floats), dim1 = rows.
// Tile = KC x TN, 4-byte elements, LDS padding: 2 dwords after every 64 dwords
// (pad_interval=5 -> 64 dwords, pad_amount=1 -> 2 dwords) => LDS row stride 66.
__device__ __forceinline__ void tdm_load_tile(const float* gsrc, unsigned lds_off,
                                              unsigned rows_valid) {
  unsigned long long ga = (unsigned long long)(uintptr_t)gsrc;
  u32x4 g0;
  g0[0] = 1u;                                        // count=1 (valid), user mode
  g0[1] = lds_off;                                   // lds_addr
  g0[2] = (unsigned)(ga & 0xFFFFFFFFu);              // global_addr[31:0]
  g0[3] = (unsigned)((ga >> 32) & 0x1FFFFFFu)        // global_addr[56:32]
          | (2u << 30);                              // type = 2 ("image")
  unsigned dim0 = (unsigned)KC;                      // no OOB along K
  unsigned dim1 = rows_valid;                        // rows beyond -> zero-fill
  i32x8 g1;
  g1[0] = (int)((2u << 16)      // data_size = 4 bytes
              | (1u << 20)      // pad_enable
              | (5u << 22)      // pad_interval: 64 dwords
              | (1u << 25));    // pad_amount: 2 dwords
  g1[1] = (int)((dim0 & 0xFFFFu) << 16);                               // dim0 lo
  g1[2] = (int)(((dim0 >> 16) & 0xFFFFu) | ((dim1 & 0xFFFFu) << 16));  // dim0 hi | dim1 lo
  g1[3] = (int)(((dim1 >> 16) & 0xFFFFu) | ((unsigned)KC << 16));      // dim1 hi | tile_dim0
  g1[4] = (int)(unsigned)TN;                                           // tile_dim1 | tile_dim2=0
  g1[5] = (int)(unsigned)FEAT;                                         // tensor_dim0_stride lo
  g1[6] = 0;                                                           // stride hi | dim1_stride lo
  g1[7] = 0;
  i32x4 z4; z4[0] = 0; z4[1] = 0; z4[2] = 0; z4[3] = 0;
  i32x8 z8; z8[0] = 0; z8[1] = 0; z8[2] = 0; z8[3] = 0;
  z8[4] = 0; z8[5] = 0; z8[6] = 0; z8[7] = 0;
  __builtin_amdgcn_tensor_load_to_lds(g0, g1, z4, z4, z8, 0);
}

// ---------------------------------------------------------------- prep: x_sq + min_loss
__global__ __launch_bounds__(FEAT) void prep_kernel(const float* __restrict__ x,
                                                    const int* __restrict__ labels,
                                                    const float* __restrict__ Mem,
                                                    float* __restrict__ xsq,
                                                    float* __restrict__ minloss,
                                                    int M) {
  __shared__ float s1[FEAT], s2[FEAT], s3[FEAT];
  int b = blockIdx.x, t = threadIdx.x;
  int lab = labels[b];
  float xv = x[(size_t)b * FEAT + t];
  float mv = Mem[(size_t)lab * FEAT + t];
  s1[t] = xv * xv; s2[t] = mv * mv; s3[t] = xv * mv;
  __syncthreads();
  for (int o = FEAT / 2; o > 0; o >>= 1) {
    if (t < o) { s1[t] += s1[t + o]; s2[t] += s2[t + o]; s3[t] += s3[t + o]; }
    __syncthreads();
  }
  if (t == 0) {
    xsq[b] = s1[0];
    float d = s1[0] + s2[0] - 2.0f * s3[0];
    d = fminf(fmaxf(d, 1e-12f), 1e12f);
    minloss[b] = d + (float)(M - 1) * 1e-12f;   // clip() applied to masked zeros too
  }
}

// ---------------------------------------------------------------- GEMM + distance
__global__ __launch_bounds__(256) void dist_kernel(const float* __restrict__ x,
                                                   const int* __restrict__ labels,
                                                   const float* __restrict__ Mem,
                                                   const float* __restrict__ xsq,
                                                   float* __restrict__ dist,
                                                   int M) {
  __shared__ float mt[2][TN * RS];
  __shared__ float msq[TN];
  const int tid  = threadIdx.x;
  const int wid  = tid >> 5;
  const int lane = tid & 31;
  const int hlf  = lane >> 4;        // 0: K 0,1 / N,M lanes 0-15; 1: K 2,3
  const int l15  = lane & 15;
  const int m0   = blockIdx.x * TN;
  const int remaining = M - m0;
  const unsigned rows = (unsigned)(remaining < TN ? remaining : TN);

  v8f acc[8];
#pragma unroll
  for (int i = 0; i < 8; ++i)
#pragma unroll
    for (int j = 0; j < 8; ++j) acc[i][j] = 0.0f;

  float msq_acc = 0.0f;
  const int brow = wid * 16 + l15;                 // A-matrix row (query index)
  const float* xrow = x + (size_t)brow * FEAT;
  const int kofs = hlf * 2;

  if (wid == 0)
    tdm_load_tile(Mem + (size_t)m0 * FEAT, (unsigned)(uintptr_t)&mt[0][0], rows);

  for (int c = 0; c < NCHUNK; ++c) {
    if (wid == 0) __builtin_amdgcn_s_wait_tensorcnt(0);   // chunk c landed in LDS
    __syncthreads();                                      // publish; prior reads done
    if (wid == 0 && c + 1 < NCHUNK)
      tdm_load_tile(Mem + (size_t)m0 * FEAT + (c + 1) * KC,
                    (unsigned)(uintptr_t)&mt[(c + 1) & 1][0], rows);
    const float* buf = mt[c & 1];

    if (tid < TN) {                                       // ||m||^2 partial
      const float* r = buf + tid * RS;
#pragma unroll
      for (int j = 0; j < KC; ++j) { float v = r[j]; msq_acc += v * v; }
    }

#pragma unroll 4
    for (int kk = 0; kk < KC; kk += 4) {
      v2f a = *(const v2f*)(xrow + c * KC + kk + kofs);   // A frag (L2-resident x)
#pragma unroll
      for (int nt = 0; nt < 8; ++nt) {
        const float* bp = buf + (nt * 16 + l15) * RS + kk + kofs;
        v2f bfrag = *(const v2f*)bp;                      // conflict-free: stride 66
        acc[nt] = __builtin_amdgcn_wmma_f32_16x16x4_f32(
            false, a, false, bfrag, (short)0, acc[nt], false, false);
      }
    }
  }
  if (tid < TN) msq[tid] = msq_acc;
  __syncthreads();

  float xs[8]; int lb[8];
#pragma unroll
  for (int v = 0; v < 8; ++v) {
    int b = wid * 16 + hlf * 8 + v;                       // C layout: M = v (+8 hi half)
    xs[v] = xsq[b]; lb[v] = labels[b];
  }
#pragma unroll
  for (int nt = 0; nt < 8; ++nt) {
    int m = m0 + nt * 16 + l15;                           // C layout: N = lane&15
    if (m < M) {
      float ms = msq[nt * 16 + l15];
#pragma unroll
      for (int v = 0; v < 8; ++v) {
        int b = wid * 16 + hlf * 8 + v;
        float d = xs[v] + ms - 2.0f * acc[nt][v];
        if (lb[v] == m) d = 0.0f;                         // dist2 = distmat*(1-mask)
        dist[(size_t)b * M + m] = d;
      }
    }
  }
}

// ---------------------------------------------------------------- binning helpers
__device__ __forceinline__ int bin_of(float d) {
  int b = (int)d;                                         // width-1.0 linear bins
  b = b < 0 ? 0 : b;
  return b > (NBINS - 1) ? (NBINS - 1) : b;
}

__global__ __launch_bounds__(256) void hist_kernel(const float* __restrict__ dist,
                                                   unsigned* __restrict__ ghist, int M) {
  __shared__ unsigned h[NBINS];
  int row = blockIdx.y, tid = threadIdx.x;
  for (int i = tid; i < NBINS; i += 256) h[i] = 0;
  __syncthreads();
  int start = blockIdx.x * HCHUNK;
  int end = start + HCHUNK; end = end < M ? end : M;
  const float* dr = dist + (size_t)row * M;
  for (int i = start + tid; i < end; i += 256) atomicAdd(&h[bin_of(dr[i])], 1u);
  __syncthreads();
  for (int i = tid; i < NBINS; i += 256)
    if (h[i]) atomicAdd(&ghist[(size_t)row * NBINS + i], h[i]);
}

__global__ void thresh_kernel(const unsigned* __restrict__ ghist,
                              const int* __restrict__ topk,
                              int* __restrict__ thr, int* __restrict__ need) {
  int b = threadIdx.x;
  unsigned k = (unsigned)topk[0];
  const unsigned* h = ghist + (size_t)b * NBINS;
  unsigned cum = 0; int t = NBINS - 1; unsigned below = 0;
  for (int i = 0; i < NBINS; ++i) {
    unsigned c = h[i];
    if (cum + c >= k) { t = i; below = cum; break; }
    cum += c;
    if (i == NBINS - 1) below = cum - c;                  // never reached (degenerate)
  }
  thr[b] = t;
  need[b] = (int)(k - below);
}

__global__ __launch_bounds__(256) void collect_kernel(const float* __restrict__ dist,
                                                      const int* __restrict__ thr,
                                                      float* __restrict__ sum_below,
                                                      float* __restrict__ bbuf,
                                                      unsigned* __restrict__ bcount,
                                                      int M) {
  __shared__ float red[256];
  int row = blockIdx.y, tid = threadIdx.x;
  int t = thr[row];
  int start = blockIdx.x * HCHUNK;
  int end = start + HCHUNK; end = end < M ? end : M;
  const float* dr = dist + (size_t)row * M;
  float local = 0.0f;
  for (int i = start + tid; i < end; i += 256) {
    float d = dr[i];
    int b = bin_of(d);
    if (b < t) local += d;
    else if (b == t) {
      unsigned idx = atomicAdd(&bcount[row], 1u);
      if (idx < CAP) bbuf[(size_t)row * CAP + idx] = d;
    }
  }
  red[tid] = local; __syncthreads();
  for (int o = 128; o > 0; o >>= 1) { if (tid < o) red[tid] += red[tid + o]; __syncthreads(); }
  if (tid == 0) atomicAdd(&sum_below[row], red[0]);
}

__global__ __launch_bounds__(256) void final_kernel(const float* __restrict__ minloss,
                                                    const float* __restrict__ sum_below,
                                                    const float* __restrict__ bbuf,
                                                    const unsigned* __restrict__ bcount,
                                                    const int* __restrict__ need,
                                                    const int* __restrict__ topk,
                                                    float* __restrict__ out, int Bn) {
  __shared__ float v[CAP];
  __shared__ float red[256];
  int row = blockIdx.x, tid = threadIdx.x;
  int n = (int)bcount[row]; n = n > CAP ? CAP : n;
  for (int i = tid; i < CAP; i += 256) v[i] = (i < n) ? bbuf[(size_t)row * CAP + i] : 3.0e38f;
  __syncthreads();
  for (int k = 2; k <= CAP; k <<= 1)
    for (int j = k >> 1; j > 0; j >>= 1) {
      for (int i = tid; i < CAP; i += 256) {
        int ij = i ^ j;
        if (ij > i) {
          bool up = ((i & k) == 0);
          float a = v[i], b2 = v[ij];
          if ((a > b2) == up) { v[i] = b2; v[ij] = a; }
        }
      }
      __syncthreads();
    }
  int r = need[row]; r = r > n ? n : r; r = r < 0 ? 0 : r;
  float s = 0.0f;
  for (int i = tid; i < r; i += 256) s += v[i];
  red[tid] = s; __syncthreads();
  for (int o = 128; o > 0; o >>= 1) { if (tid < o) red[tid] += red[tid + o]; __syncthreads(); }
  if (tid == 0) {
    float tk = (float)topk[0];
    float ml = (sum_below[row] + red[0]) / (tk - 1.0f);
    float mn = minloss[row];
    float ls = fmaxf(mn - ml + 1.0f, 0.0f);
    float inv = 1.0f / (float)Bn;
    atomicAdd(&out[0], ls * inv);
    atomicAdd(&out[1], mn * inv);
    atomicAdd(&out[2], ml * inv);
  }
}

// ---------------------------------------------------------------- launch
extern "C" void kernel_launch(void* const* d_in, const int* in_sizes, int n_in,
                              void* d_out, int out_size, void* d_ws, size_t ws_size,
                              hipStream_t stream) {
  const float* x      = (const float*)d_in[0];
  const int*   labels = (const int*)d_in[1];
  const float* Mem    = (const float*)d_in[2];
  const int*   topk   = (const int*)d_in[3];
  const int Bn = in_sizes[1];               // 128
  const int D  = in_sizes[0] / Bn;          // 256
  const int M  = in_sizes[2] / D;           // 500000
  float* out = (float*)d_out;

  char* w = (char*)d_ws;
  auto carve = [&](size_t bytes) { char* p = w; w += (bytes + 255) & ~(size_t)255; return p; };
  float*    dist      = (float*)carve((size_t)Bn * M * sizeof(float));
  unsigned* ghist     = (unsigned*)carve((size_t)Bn * NBINS * sizeof(unsigned));
  float*    xsq       = (float*)carve(Bn * sizeof(float));
  float*    minloss   = (float*)carve(Bn * sizeof(float));
  int*      thr       = (int*)carve(Bn * sizeof(int));
  int*      need      = (int*)carve(Bn * sizeof(int));
  float*    sum_below = (float*)carve(Bn * sizeof(float));
  unsigned* bcount    = (unsigned*)carve(Bn * sizeof(unsigned));
  float*    bbuf      = (float*)carve((size_t)Bn * CAP * sizeof(float));

  (void)hipMemsetAsync(ghist, 0, (size_t)Bn * NBINS * sizeof(unsigned), stream);
  (void)hipMemsetAsync(sum_below, 0, Bn * sizeof(float), stream);
  (void)hipMemsetAsync(bcount, 0, Bn * sizeof(unsigned), stream);
  (void)hipMemsetAsync(out, 0, 3 * sizeof(float), stream);

  prep_kernel<<<Bn, FEAT, 0, stream>>>(x, labels, Mem, xsq, minloss, M);
  dist_kernel<<<(M + TN - 1) / TN, 256, 0, stream>>>(x, labels, Mem, xsq, dist, M);
  int ch = (M + HCHUNK - 1) / HCHUNK;
  hist_kernel<<<dim3(ch, Bn), 256, 0, stream>>>(dist, ghist, M);
  thresh_kernel<<<1, Bn, 0, stream>>>(ghist, topk, thr, need);
  collect_kernel<<<dim3(ch, Bn), 256, 0, stream>>>(dist, thr, sum_below, bbuf, bcount, M);
  final_kernel<<<Bn, 256, 0, stream>>>(minloss, sum_below, bbuf, bcount, need, topk, out, Bn);
}